// TripPhraseDistRelation_40750649704567
// MI455X (gfx1250) — compile-verified
//
#include <hip/hip_runtime.h>

typedef __attribute__((ext_vector_type(2))) float v2f;
typedef __attribute__((ext_vector_type(4))) float v4f;
typedef __attribute__((ext_vector_type(8))) float v8f;
typedef __attribute__((ext_vector_type(4))) int   v4i;

// Address-space-qualified pointer types for the async-to-LDS builtin
// (param 0 is AS1 "global/device", param 1 is AS3 "LDS").
typedef __attribute__((address_space(1))) v4i* gptr_v4i;
typedef __attribute__((address_space(3))) v4i* lptr_v4i;

#define B_    2
#define L_    64
#define F_    128
#define PROWS 65                   // L_+1 prefix rows
#define PQ    (PROWS * (F_ / 4))   // 2080 float4 per batch slab

// ---------------------------------------------------------------------------
// Kernel A: exclusive prefix sums P[b,t,f] via triangular matmul on WMMA.
//   P[1+i, :] = sum_{l<=i} x[l, :]  ==  Tri(64x64, Tri[i,l]=(l<=i)) @ X(64x128)
// Tiled 16x16x4 f32 WMMA; 8 blocks = (b,ti), 8 waves/block = tj.
// ---------------------------------------------------------------------------
__global__ __launch_bounds__(256) void cumsum_wmma_kernel(const float* __restrict__ x,
                                                          float* __restrict__ P) {
    const int bx   = blockIdx.x;        // 0..7
    const int b    = bx >> 2;
    const int ti   = bx & 3;            // output row tile (16 rows)
    const int tid  = threadIdx.x;
    const int lane = tid & 31;
    const int tj   = tid >> 5;          // column tile (16 cols), one wave each
    const int half = lane >> 4;         // lane half selects K pair
    const int kb   = half * 2;
    const int rowg = ti * 16 + (lane & 15);   // A-matrix global row (M)
    const int col  = tj * 16 + (lane & 15);   // B/D column (N)

    v8f acc = {};
    #pragma unroll
    for (int k0 = 0; k0 < L_; k0 += 4) {
        // A 16x4 layout: VGPR0 = K{0,2}, VGPR1 = K{1,3} across lane halves
        v2f a, bb;
        a.x = ((k0 + kb)     <= rowg) ? 1.0f : 0.0f;
        a.y = ((k0 + kb + 1) <= rowg) ? 1.0f : 0.0f;
        // B 4x16 layout: row striped across lanes, same K pairing
        bb.x = x[(b * L_ + (k0 + kb))     * F_ + col];
        bb.y = x[(b * L_ + (k0 + kb + 1)) * F_ + col];
        acc = __builtin_amdgcn_wmma_f32_16x16x4_f32(
                  /*neg_a=*/false, a, /*neg_b=*/false, bb,
                  /*c_mod=*/(short)0, acc, /*reuse_a=*/false, /*reuse_b=*/false);
    }
    // D layout: VGPR v -> M = v + 8*half, N = lane&15
    #pragma unroll
    for (int v = 0; v < 8; ++v) {
        const int row = ti * 16 + v + half * 8;          // Tri row i -> P row i+1
        P[(b * PROWS + row + 1) * F_ + col] = acc[v];
    }
    if (ti == 0) P[(b * PROWS + 0) * F_ + col] = 0.0f;   // P[0,:] = 0
}

// ---------------------------------------------------------------------------
// Kernel B: streaming broadcast-store of the (B,L,L,L,2F) tensor.
// Grid: 1024 blocks = (b, tr, sp-chunk-of-8). Stage P[b] (33 KB) in LDS via
// async-to-LDS, then each thread emits one NT b128 store per iteration.
// ---------------------------------------------------------------------------
__global__ __launch_bounds__(256) void dist_store_kernel(const float* __restrict__ P,
                                                         float* __restrict__ out) {
    const int bx  = blockIdx.x;                 // 0..1023
    const int sp0 = (bx & 7) * 8;
    const int tr  = (bx >> 3) & 63;
    const int b   = bx >> 9;
    const int tid = threadIdx.x;

    __shared__ v4f   sP4[PQ];                   // 33,280 B: P[b] slab
    __shared__ float sInv[L_];                  // 1/max(d,1) table

    const v4f* Pg4 = (const v4f*)(P + b * PROWS * F_);

#if __has_builtin(__builtin_amdgcn_global_load_async_to_lds_b128)
    #pragma unroll
    for (int itr = 0; itr < 9; ++itr) {
        const int idx = itr * 256 + tid;
        if (idx < PQ) {
            __builtin_amdgcn_global_load_async_to_lds_b128(
                (gptr_v4i)(Pg4 + idx),
                (lptr_v4i)(sP4 + idx),
                /*offset=*/0, /*cpol=*/0);
        }
    }
    if (tid < L_) sInv[tid] = 1.0f / (float)(tid ? tid : 1);
  #if __has_builtin(__builtin_amdgcn_s_wait_asynccnt)
    __builtin_amdgcn_s_wait_asynccnt(0);
  #else
    asm volatile("s_wait_asynccnt 0" ::: "memory");
  #endif
#else
    #pragma unroll
    for (int itr = 0; itr < 9; ++itr) {
        const int idx = itr * 256 + tid;
        if (idx < PQ) sP4[idx] = Pg4[idx];
    }
    if (tid < L_) sInv[tid] = 1.0f / (float)(tid ? tid : 1);
#endif
    __syncthreads();

    // Thread -> (row-quarter, float4 slot). Wave = 512 B contiguous store.
    const int  f4        = tid & 63;      // float4 index within 256-ch row
    const int  rq        = tid >> 6;      // which of 4 rows this iteration
    const bool firstHalf = (f4 < 32);     // channels [0,128) vs [128,256)
    const int  q         = firstHalf ? f4 : (f4 - 32);
    v4f* out4 = (v4f*)out;
    const size_t c0 = ((size_t)b << 18) | ((size_t)tr << 6);

    #pragma unroll 4
    for (int it = 0; it < 128; ++it) {
        const int pair = it * 4 + rq;                // 0..511
        const int lm   = pair >> 3;
        const int sp   = sp0 + (pair & 7);
        const int i    = firstHalf ? lm : sp;        // M is symmetric in (i,tr)
        const int lo   = (i < tr) ? i : tr;
        const int hi   = (i < tr) ? tr : i;
        v4f val = (sP4[hi * 32 + q] - sP4[lo * 32 + q]) * sInv[hi - lo];
        const float msk = (lm != tr && lm != sp && sp != tr) ? 1.0f : 0.0f;
        val *= msk;
        const size_t o = ((c0 + ((size_t)lm << 12) + (size_t)sp) << 6) + (size_t)f4;
        __builtin_nontemporal_store(val, out4 + o);
    }
}

extern "C" void kernel_launch(void* const* d_in, const int* in_sizes, int n_in,
                              void* d_out, int out_size, void* d_ws, size_t ws_size,
                              hipStream_t stream) {
    const float* x   = (const float*)d_in[0];
    float*       P   = (float*)d_ws;              // needs 2*65*128*4 = 66,560 B
    float*       out = (float*)d_out;
    (void)in_sizes; (void)n_in; (void)out_size; (void)ws_size;

    cumsum_wmma_kernel<<<8, 256, 0, stream>>>(x, P);
    dist_store_kernel<<<1024, 256, 0, stream>>>(P, out);
}